// DBISSF_SS_11330123727074
// MI455X (gfx1250) — compile-verified
//
#include <hip/hip_runtime.h>
#include <hip/hip_bf16.h>

// Sizes from the reference
#define D_INNER 192
#define D_STATE 16
#define DT_RANK 6
#define B_      4
#define L_      4096
#define M_      (B_ * L_)          // 16384 rows
#define KPROJ   38                 // DT_RANK + 2*D_STATE
#define KPAD    48                 // padded to 3 tiles of 16
#define SLOT    ((size_t)M_ * D_INNER)  // one output tensor, flat

typedef __attribute__((ext_vector_type(2))) float v2f;
typedef __attribute__((ext_vector_type(8))) float v8f;

// ---------------------------------------------------------------------------
// x_share = x_share_rgb + x_share_e
__global__ __launch_bounds__(256) void add_kernel(const float* __restrict__ a,
                                                  const float* __restrict__ b,
                                                  float* __restrict__ c, int n) {
    int i = blockIdx.x * blockDim.x + threadIdx.x;
    if (i < n) c[i] = a[i] + b[i];
}

// A = -exp(Alog) for the three branches -> Aws[3][192][16]
__global__ __launch_bounds__(256) void aexp_kernel(const float* __restrict__ A1,
                                                   const float* __restrict__ A2,
                                                   const float* __restrict__ As,
                                                   float* __restrict__ Aws) {
    int i = blockIdx.x * blockDim.x + threadIdx.x;
    const int per = D_INNER * D_STATE;
    if (i < 3 * per) {
        int t = i / per, r = i % per;
        const float* src = (t == 0) ? A1 : (t == 1) ? A2 : As;
        Aws[i] = -__expf(src[r]);
    }
}

// ---------------------------------------------------------------------------
// x_dbl[t][m][k'] = sum_d X_t[m][d] * W_t[k'][d]  via V_WMMA_F32_16X16X4_F32.
// One wave per (branch t, mtile, ntile); 3 * 1024 * 3 = 9216 waves.
// Branchless: out-of-range n columns clamp the W row pointer to row 0; the
// resulting garbage lands only in padded cols 38..47 which are never read.
// Two accumulators break the WMMA RAW chain for 2x matrix-pipe ILP.
__global__ __launch_bounds__(256) void proj_wmma_kernel(
    const float* __restrict__ Xrgb, const float* __restrict__ Xe,
    const float* __restrict__ Xsh,
    const float* __restrict__ W1, const float* __restrict__ W2,
    const float* __restrict__ Ws,
    float* __restrict__ xdbl)   // [3][M_][KPAD]
{
    int wave = (blockIdx.x * blockDim.x + threadIdx.x) >> 5;
    int lane = threadIdx.x & 31;
    int nt = wave % 3;
    int mt = (wave / 3) % (M_ / 16);
    int t  = wave / (3 * (M_ / 16));

    const float* X = (t == 0) ? Xrgb : (t == 1) ? Xe : Xsh;
    const float* W = (t == 0) ? W1 : (t == 1) ? W2 : Ws;

    const int m0 = mt * 16;
    const int n0 = nt * 16;
    const int hf  = lane >> 4;   // 0: lanes 0-15 (K 0,1), 1: lanes 16-31 (K 2,3)
    const int lid = lane & 15;

    // A: lane holds row (m0+lid), K pair (2*hf, 2*hf+1)
    const float* arow = X + (size_t)(m0 + lid) * D_INNER + 2 * hf;
    // B = W^T: lane holds column ncol (clamped), K pair (2*hf, 2*hf+1)
    const int ncol = n0 + lid;
    const int ncol_c = (ncol < KPROJ) ? ncol : 0;   // dead padded cols
    const float* wrow = W + (size_t)ncol_c * D_INNER + 2 * hf;

    v8f acc0 = {0.f, 0.f, 0.f, 0.f, 0.f, 0.f, 0.f, 0.f};
    v8f acc1 = {0.f, 0.f, 0.f, 0.f, 0.f, 0.f, 0.f, 0.f};
    #pragma unroll
    for (int k0 = 0; k0 < D_INNER; k0 += 8) {
        v2f a0 = *(const v2f*)(arow + k0);
        v2f b0 = *(const v2f*)(wrow + k0);
        v2f a1 = *(const v2f*)(arow + k0 + 4);
        v2f b1 = *(const v2f*)(wrow + k0 + 4);
        acc0 = __builtin_amdgcn_wmma_f32_16x16x4_f32(
            false, a0, false, b0, (short)0, acc0, false, false);
        acc1 = __builtin_amdgcn_wmma_f32_16x16x4_f32(
            false, a1, false, b1, (short)0, acc1, false, false);
    }

    // D layout: VGPR v -> row v (lanes 0-15) / row 8+v (lanes 16-31), col = lid
    size_t base = ((size_t)t * M_ + m0) * KPAD;
    #pragma unroll
    for (int v = 0; v < 8; ++v) {
        int row = v + 8 * hf;
        xdbl[base + (size_t)row * KPAD + n0 + lid] = acc0[v] + acc1[v];
    }
}

// ---------------------------------------------------------------------------
// delta[t][m][d] = softplus( sum_r xdbl[t][m][r] * Wdt_t[d][r] + bdt_t[d] )
__global__ __launch_bounds__(256) void dt_kernel(
    const float* __restrict__ xdbl,
    const float* __restrict__ Wdt1, const float* __restrict__ Wdt2,
    const float* __restrict__ Wdts,
    const float* __restrict__ bdt1, const float* __restrict__ bdt2,
    const float* __restrict__ bdts,
    float* __restrict__ delta)  // [3][M_][192]
{
    int idx = blockIdx.x * blockDim.x + threadIdx.x;
    const int total = 3 * M_ * D_INNER;
    if (idx >= total) return;
    int d  = idx % D_INNER;
    int ml = (idx / D_INNER) % M_;
    int t  = idx / (D_INNER * M_);
    const float* Wdt = (t == 0) ? Wdt1 : (t == 1) ? Wdt2 : Wdts;
    const float* bdt = (t == 0) ? bdt1 : (t == 1) ? bdt2 : bdts;
    const float* row = xdbl + ((size_t)t * M_ + ml) * KPAD;
    const float* w   = Wdt + d * DT_RANK;
    float s = bdt[d];
    #pragma unroll
    for (int r = 0; r < DT_RANK; ++r) s = fmaf(row[r], w[r], s);
    // stable softplus
    delta[idx] = (s > 20.f) ? s : __logf(1.f + __expf(s));
}

// ---------------------------------------------------------------------------
// 4 selective scans. One lane per (scan, b, d, n): 4*4*192*16 = 49152 lanes.
// Writes y + D*u directly into d_out at the right slot, [b][l][d] layout.
__global__ __launch_bounds__(256) void scan_kernel(
    const float* __restrict__ Xrgb, const float* __restrict__ Xe,
    const float* __restrict__ xdbl, const float* __restrict__ delta,
    const float* __restrict__ Aws,
    const float* __restrict__ D1, const float* __restrict__ D2,
    const float* __restrict__ Ds,
    float* __restrict__ out)
{
    int tid = blockIdx.x * blockDim.x + threadIdx.x;
    int n  = tid & 15;
    int cg = tid >> 4;                  // 0..3071
    int scan = cg / (B_ * D_INNER);
    int rem  = cg % (B_ * D_INNER);
    int b = rem / D_INNER;
    int d = rem % D_INNER;

    // scan 0: (u_rgb, d0, A0, B0, C1, D1) -> y_rgb     (out slot 0)
    // scan 1: (u_e,   d1, A1, B1, C0, D2) -> y_e       (out slot 2)
    // scan 2: (u_rgb, d2, A2, B2, C2, Ds) -> y_sh_rgb  (out slot 1)
    // scan 3: (u_e,   d2, A2, B2, C2, Ds) -> y_sh_e    (out slot 3)
    const float* u  = (scan == 0 || scan == 2) ? Xrgb : Xe;
    int br = (scan == 0) ? 0 : (scan == 1) ? 1 : 2;   // delta / A / B branch
    int Cb = (scan == 0) ? 1 : (scan == 1) ? 0 : 2;   // C branch
    const float* Dv = (scan == 0) ? D1 : (scan == 1) ? D2 : Ds;
    size_t outoff = (scan == 0) ? 0 : (scan == 1) ? 2 * SLOT
                  : (scan == 2) ? SLOT : 3 * SLOT;

    const float A_ln = Aws[((size_t)br * D_INNER + d) * D_STATE + n];
    const float Dd   = Dv[d];

    const float* dptr = delta + ((size_t)br * M_ + (size_t)b * L_) * D_INNER + d;
    const float* uptr = u + ((size_t)b * L_) * D_INNER + d;
    const float* Bptr = xdbl + ((size_t)br * M_ + (size_t)b * L_) * KPAD + DT_RANK + n;
    const float* Cptr = xdbl + ((size_t)Cb * M_ + (size_t)b * L_) * KPAD + DT_RANK + D_STATE + n;
    float* optr = out + outoff + ((size_t)b * L_) * D_INNER + d;

    float h = 0.f;
    for (int l = 0; l < L_; ++l) {
        // speculative prefetch 4 steps ahead (global_prefetch_b8; counter-free)
        __builtin_prefetch(dptr + (size_t)(l + 4) * D_INNER, 0, 1);
        __builtin_prefetch(uptr + (size_t)(l + 4) * D_INNER, 0, 1);
        __builtin_prefetch(Bptr + (size_t)(l + 4) * KPAD, 0, 1);
        __builtin_prefetch(Cptr + (size_t)(l + 4) * KPAD, 0, 1);

        float dt = dptr[(size_t)l * D_INNER];
        float ut = uptr[(size_t)l * D_INNER];
        float Bt = Bptr[(size_t)l * KPAD];
        float Ct = Cptr[(size_t)l * KPAD];
        float dA = __expf(dt * A_ln);
        h = fmaf(dA, h, dt * ut * Bt);
        float p = h * Ct;
        // sum over the 16 state lanes of this channel
        p += __shfl_xor(p, 1, 16);
        p += __shfl_xor(p, 2, 16);
        p += __shfl_xor(p, 4, 16);
        p += __shfl_xor(p, 8, 16);
        if (n == 0) optr[(size_t)l * D_INNER] = fmaf(Dd, ut, p);
    }
}

// ---------------------------------------------------------------------------
// In-place layernorm over rows of 192; one wave per row; 4*16384 rows.
__global__ __launch_bounds__(256) void ln_kernel(
    float* __restrict__ out,
    const float* __restrict__ g1, const float* __restrict__ be1,
    const float* __restrict__ g2, const float* __restrict__ be2,
    const float* __restrict__ gs, const float* __restrict__ bes)
{
    int wave = (blockIdx.x * blockDim.x + threadIdx.x) >> 5;
    int lane = threadIdx.x & 31;
    int slot = wave / M_;          // output slot 0..3
    int row  = wave % M_;
    const float* g; const float* be;
    if (slot == 0)      { g = g1; be = be1; }
    else if (slot == 2) { g = g2; be = be2; }
    else                { g = gs; be = bes; }

    float* p = out + ((size_t)slot * M_ + row) * D_INNER;
    float x[6];
    float s = 0.f, s2 = 0.f;
    #pragma unroll
    for (int i = 0; i < 6; ++i) {
        x[i] = p[lane + 32 * i];
        s  += x[i];
        s2 += x[i] * x[i];
    }
    #pragma unroll
    for (int m = 1; m < 32; m <<= 1) {
        s  += __shfl_xor(s,  m, 32);
        s2 += __shfl_xor(s2, m, 32);
    }
    const float mu  = s * (1.f / D_INNER);
    const float var = s2 * (1.f / D_INNER) - mu * mu;
    const float inv = rsqrtf(var + 1e-5f);
    #pragma unroll
    for (int i = 0; i < 6; ++i) {
        int c = lane + 32 * i;
        p[c] = (x[i] - mu) * inv * g[c] + be[c];
    }
}

// ---------------------------------------------------------------------------
extern "C" void kernel_launch(void* const* d_in, const int* in_sizes, int n_in,
                              void* d_out, int out_size, void* d_ws, size_t ws_size,
                              hipStream_t stream) {
    (void)in_sizes; (void)n_in; (void)out_size; (void)ws_size;
    const float* x_rgb       = (const float*)d_in[0];
    const float* x_share_rgb = (const float*)d_in[1];
    const float* x_e         = (const float*)d_in[2];
    const float* x_share_e   = (const float*)d_in[3];
    const float* Wx1  = (const float*)d_in[4];
    const float* Wx2  = (const float*)d_in[5];
    const float* Wxs  = (const float*)d_in[6];
    const float* Wdt1 = (const float*)d_in[7];
    const float* Wdt2 = (const float*)d_in[8];
    const float* Wdts = (const float*)d_in[9];
    const float* bdt1 = (const float*)d_in[10];
    const float* bdt2 = (const float*)d_in[11];
    const float* bdts = (const float*)d_in[12];
    const float* Alog1 = (const float*)d_in[13];
    const float* Alog2 = (const float*)d_in[14];
    const float* Alogs = (const float*)d_in[15];
    const float* D1 = (const float*)d_in[16];
    const float* D2 = (const float*)d_in[17];
    const float* Ds = (const float*)d_in[18];
    const float* g1  = (const float*)d_in[19];
    const float* be1 = (const float*)d_in[20];
    const float* g2  = (const float*)d_in[21];
    const float* be2 = (const float*)d_in[22];
    const float* gs  = (const float*)d_in[23];
    const float* bes = (const float*)d_in[24];

    float* ws = (float*)d_ws;
    const size_t XSHARE_N = (size_t)M_ * D_INNER;        // 3,145,728
    const size_t XDBL_N   = (size_t)3 * M_ * KPAD;       // 2,359,296
    const size_t DELTA_N  = (size_t)3 * M_ * D_INNER;    // 9,437,184
    float* xshare = ws;
    float* xdbl   = ws + XSHARE_N;
    float* delta  = ws + XSHARE_N + XDBL_N;
    float* Aws    = ws + XSHARE_N + XDBL_N + DELTA_N;
    float* out    = (float*)d_out;

    add_kernel<<<(int)((XSHARE_N + 255) / 256), 256, 0, stream>>>(
        x_share_rgb, x_share_e, xshare, (int)XSHARE_N);
    aexp_kernel<<<(3 * D_INNER * D_STATE + 255) / 256, 256, 0, stream>>>(
        Alog1, Alog2, Alogs, Aws);
    // 9216 waves = 1152 blocks * 8 waves
    proj_wmma_kernel<<<1152, 256, 0, stream>>>(
        x_rgb, x_e, xshare, Wx1, Wx2, Wxs, xdbl);
    dt_kernel<<<(int)((DELTA_N + 255) / 256), 256, 0, stream>>>(
        xdbl, Wdt1, Wdt2, Wdts, bdt1, bdt2, bdts, delta);
    // 49152 lanes = 192 blocks * 256
    scan_kernel<<<192, 256, 0, stream>>>(
        x_rgb, x_e, xdbl, delta, Aws, D1, D2, Ds, out);
    // 65536 waves = 8192 blocks * 8 waves
    ln_kernel<<<8192, 256, 0, stream>>>(out, g1, be1, g2, be2, gs, bes);
}